// STN_89172110999959
// MI455X (gfx1250) — compile-verified
//
#include <hip/hip_runtime.h>
#include <hip/hip_bf16.h>

// STN bilinear sampler, MI455X (gfx1250, wave32).
//
// B=16, H=W=512, C=8, fp32, DOWNSAMPLE_FACTOR=1.
// Roofline: 128 MiB image in + 128 MiB out => ~11 us floor at 23.3 TB/s.
// ~0.17 GFLOP of arithmetic -> pure bandwidth/gather kernel; WMMA inapplicable.
// The whole image fits in the 192 MB L2, so the 4 bilinear taps are L2 hits
// after first touch. Strategy:
//   - per-block-uniform (b, oy) decode -> theta via s_load, affine row-terms on SALU
//   - B128 vector path: 4 taps x 2 global_load_b128, 2 global_store_b128 stores
//   - linear global_prefetch sweep warms L2 for the data-dependent gathers
//   - non-temporal global_store_b128 so the streamed output can't evict the
//     L2-resident image

typedef float v4f __attribute__((ext_vector_type(4)));

#define IM_H 512
#define IM_W 512
#define IM_C 8

__global__ __launch_bounds__(256) void stn_bilinear_kernel(
    const float* __restrict__ im,     // (16, 512, 512, 8)
    const float* __restrict__ theta,  // (16, 6)
    float* __restrict__ out)          // (16, 512, 512, 8)
{
    // Grid decode: 2 blocks per row (256 px each), 512 rows, 16 batches.
    // b and oy depend only on blockIdx.x -> SGPR-uniform -> theta via s_load.
    const int bid = blockIdx.x;
    const int b   = bid >> 10;                       // 1024 blocks per batch
    const int oy  = (bid >> 1) & (IM_H - 1);         // row (scalar)
    const int ox  = ((bid & 1) << 8) + threadIdx.x;  // column (per-lane)

    const size_t base = (size_t)b * (IM_H * IM_W * IM_C);
    const size_t lin  = (size_t)(oy * IM_W + ox) * IM_C;  // this thread's pixel

    // Linear L2-warming sweep: across the whole grid, every 32B chunk of the
    // input image is prefetched exactly once (gfx1250 global_prefetch path).
    // Fire-and-forget: no counter, overlaps with the data-dependent gathers.
    __builtin_prefetch(im + base + lin, 0, 3);

    const float t00 = theta[b * 6 + 0];
    const float t01 = theta[b * 6 + 1];
    const float t02 = theta[b * 6 + 2];
    const float t10 = theta[b * 6 + 3];
    const float t11 = theta[b * 6 + 4];
    const float t12 = theta[b * 6 + 5];

    // linspace(-1, 1, 512): step = 2/511
    const float step = 2.0f / 511.0f;
    const float yn = fmaf(step, (float)oy, -1.0f);   // scalar per block
    const float xn = fmaf(step, (float)ox, -1.0f);   // per lane

    // Affine transform (theta.reshape(2,3) @ [xn, yn, 1]); the yn/t02 terms
    // are block-uniform and fold onto the SALU.
    const float xs = fmaf(t00, xn, fmaf(t01, yn, t02));
    const float ys = fmaf(t10, xn, fmaf(t11, yn, t12));

    // Normalized -> pixel coords: (v + 1) * (512/2)
    const float x = (xs + 1.0f) * 256.0f;
    const float y = (ys + 1.0f) * 256.0f;

    const int x0 = (int)floorf(x);
    const int y0 = (int)floorf(y);
    const int x1 = x0 + 1;
    const int y1 = y0 + 1;

    // Reference clamps both axes to [0, H-1] (== W-1 here).
    const int x0c = min(max(x0, 0), IM_W - 1);
    const int x1c = min(max(x1, 0), IM_W - 1);
    const int y0c = min(max(y0, 0), IM_H - 1);
    const int y1c = min(max(y1, 0), IM_H - 1);

    // Weights from the CLAMPED corners (matches reference exactly).
    const float x0f = (float)x0c, x1f = (float)x1c;
    const float y0f = (float)y0c, y1f = (float)y1c;
    const float wa = (x1f - x) * (y1f - y);
    const float wb = (x1f - x) * (y - y0f);
    const float wc = (x - x0f) * (y1f - y);
    const float wd = (x - x0f) * (y - y0f);

    // Gather 4 corners x 8 channels. Each pixel's channels are 32 contiguous,
    // 32B-aligned bytes -> two global_load_b128 per tap (regular temporal:
    // we WANT these resident in L2 for the overlapping taps of neighbors).
    const v4f* pa = (const v4f*)(im + base + (size_t)(y0c * IM_W + x0c) * IM_C);
    const v4f* pb = (const v4f*)(im + base + (size_t)(y1c * IM_W + x0c) * IM_C);
    const v4f* pc = (const v4f*)(im + base + (size_t)(y0c * IM_W + x1c) * IM_C);
    const v4f* pd = (const v4f*)(im + base + (size_t)(y1c * IM_W + x1c) * IM_C);

    const v4f Ia0 = pa[0], Ia1 = pa[1];
    const v4f Ib0 = pb[0], Ib1 = pb[1];
    const v4f Ic0 = pc[0], Ic1 = pc[1];
    const v4f Id0 = pd[0], Id1 = pd[1];

    // Weighted sum (contracts to v_fma chains under -O3).
    v4f o0 = Ia0 * wa + Ib0 * wb + Ic0 * wc + Id0 * wd;
    v4f o1 = Ia1 * wa + Ib1 * wb + Ic1 * wc + Id1 * wd;

    // Streaming output: non-temporal global_store_b128 (TH=NT) so the 128 MiB
    // write-once output does not evict the L2-resident input image.
    v4f* po = (v4f*)(out + base + lin);
    __builtin_nontemporal_store(o0, po + 0);
    __builtin_nontemporal_store(o1, po + 1);
}

extern "C" void kernel_launch(void* const* d_in, const int* in_sizes, int n_in,
                              void* d_out, int out_size, void* d_ws, size_t ws_size,
                              hipStream_t stream) {
    (void)in_sizes; (void)n_in; (void)out_size; (void)d_ws; (void)ws_size;
    const float* im    = (const float*)d_in[0];  // (16, 512, 512, 8) fp32
    const float* theta = (const float*)d_in[1];  // (16, 6) fp32
    float* out         = (float*)d_out;          // (16, 512, 512, 8) fp32

    // 16 batches * 512 rows * 2 blocks/row = 16384 blocks of 256 threads
    // (8 wave32 per block on CDNA5).
    dim3 grid(16 * 512 * 2);
    dim3 block(256);
    stn_bilinear_kernel<<<grid, block, 0, stream>>>(im, theta, out);
}